// BCMPlasticity_81003083202674
// MI455X (gfx1250) — compile-verified
//
#include <hip/hip_runtime.h>
#include <stdint.h>

// ---------------------------------------------------------------------------
// BCM plasticity step, fp32-exact, V_WMMA_F32_16X16X4_F32 +
// GLOBAL_LOAD_ASYNC_TO_LDS_B128 double-buffered tile staging (CDNA5 paths).
//   out  = x @ W^T                       (8192x4096 @ 4096x4096)
//   th'  = th + (mean_b(out)^2 - th)/TAU
//   W'   = W + (LR/B) * relu(out - th')^T @ x
// ---------------------------------------------------------------------------

typedef __attribute__((ext_vector_type(2))) float v2f;
typedef __attribute__((ext_vector_type(8))) float v8f;

#define BATCH 8192
#define INF   4096   // in-features
#define OUTF  4096   // out-features

#define TILE_M 128
#define TILE_N 128
#define TILE_K 32
#define LDS_K_STRIDE 36   // [row][k] tiles: 32 data + pad; 144B rows (16B aligned),
                          // conflict-free b64 frag reads (gcd(36,64)=4 -> 16 banks/half)
#define LDS_M_STRIDE 132  // [k][m] tiles: 528B rows (16B aligned), stride-1 frag reads

// Async tile copy: 4 x b128 per lane, INST_OFFSET applies to both LDS and global
// addresses (ISA 15.18.3), so one LDS-offset VGPR + one global-offset VGPR.
#define ASYNC_TILE_B128(ldsoff, voff, sbase)                                    \
    asm volatile(                                                               \
        "global_load_async_to_lds_b128 %0, %1, %2\n\t"                          \
        "global_load_async_to_lds_b128 %0, %1, %2 offset:16\n\t"                \
        "global_load_async_to_lds_b128 %0, %1, %2 offset:32\n\t"                \
        "global_load_async_to_lds_b128 %0, %1, %2 offset:48"                    \
        :: "v"(ldsoff), "v"(voff), "s"(sbase) : "memory")

#define WAIT_ASYNC0() asm volatile("s_wait_asynccnt 0" ::: "memory")

// low 32 bits of a flat shared-aperture address == DS byte offset
__device__ __forceinline__ uint32_t lds_off(const void* p) {
    return (uint32_t)(uintptr_t)p;
}

// ------------------------- GEMM1: out = x @ W^T ---------------------------
__global__ __launch_bounds__(256)
void bcm_gemm_fwd(const float* __restrict__ X, const float* __restrict__ W,
                  float* __restrict__ Out)
{
    __shared__ float sA[2][TILE_M * LDS_K_STRIDE]; // x tile, [m][k]
    __shared__ float sB[2][TILE_N * LDS_K_STRIDE]; // W tile, [n][k]

    const int t    = threadIdx.x;
    const int m0   = blockIdx.y * TILE_M;
    const int n0   = blockIdx.x * TILE_N;
    const int wave = t >> 5;
    const int lane = t & 31;
    const int kh   = (lane >> 4) & 1;   // half-wave selector
    const int lr   = lane & 15;
    const int wm   = (wave >> 2) * 64;  // wave M offset (0 / 64)
    const int wn   = (wave & 3) * 32;   // wave N offset (0/32/64/96)

    const int lrow = t >> 1;            // 0..127
    const int lk   = (t & 1) * 16;      // 0 / 16

    const uint32_t ldsA[2] = { lds_off(&sA[0][lrow * LDS_K_STRIDE + lk]),
                               lds_off(&sA[1][lrow * LDS_K_STRIDE + lk]) };
    const uint32_t ldsB[2] = { lds_off(&sB[0][lrow * LDS_K_STRIDE + lk]),
                               lds_off(&sB[1][lrow * LDS_K_STRIDE + lk]) };
    const uint32_t voffA = (uint32_t)((m0 + lrow) * INF + lk) * 4u;  // bytes
    const uint32_t voffB = (uint32_t)((n0 + lrow) * INF + lk) * 4u;  // bytes

    v8f c[4][2];
#pragma unroll
    for (int i = 0; i < 4; ++i)
#pragma unroll
        for (int j = 0; j < 2; ++j) c[i][j] = (v8f)0.0f;

    // prologue: stage tile 0 into buffer 0
    ASYNC_TILE_B128(ldsA[0], voffA, X);
    ASYNC_TILE_B128(ldsB[0], voffB, W);
    WAIT_ASYNC0();
    __syncthreads();

    for (int k0 = 0; k0 < INF; k0 += TILE_K) {
        const int cur = (k0 / TILE_K) & 1;

        // issue next tile's async copies into the other buffer
        if (k0 + TILE_K < INF) {
            ASYNC_TILE_B128(ldsA[cur ^ 1], voffA, X + k0 + TILE_K);
            ASYNC_TILE_B128(ldsB[cur ^ 1], voffB, W + k0 + TILE_K);
        }

        const float* __restrict__ curA = sA[cur];
        const float* __restrict__ curB = sB[cur];
#pragma unroll
        for (int kk = 0; kk < TILE_K; kk += 4) {
            const int ko = kk + 2 * kh;   // A frag: VGPR0/1 = K{ko,ko+1}
            v2f a[4], b[2];
#pragma unroll
            for (int i = 0; i < 4; ++i)
                a[i] = *(const v2f*)&curA[(wm + i * 16 + lr) * LDS_K_STRIDE + ko];
#pragma unroll
            for (int j = 0; j < 2; ++j)
                b[j] = *(const v2f*)&curB[(wn + j * 16 + lr) * LDS_K_STRIDE + ko];
#pragma unroll
            for (int i = 0; i < 4; ++i)
#pragma unroll
                for (int j = 0; j < 2; ++j)
                    c[i][j] = __builtin_amdgcn_wmma_f32_16x16x4_f32(
                        false, a[i], false, b[j], (short)0, c[i][j], false, false);
        }

        WAIT_ASYNC0();      // next buffer fully written (per-wave)
        __syncthreads();    // all waves done reading cur + writes visible
    }

    // ---- store C (VGPR v: lanes0-15 -> M=v, lanes16-31 -> M=v+8) ----
#pragma unroll
    for (int i = 0; i < 4; ++i)
#pragma unroll
        for (int j = 0; j < 2; ++j)
#pragma unroll
            for (int v = 0; v < 8; ++v) {
                const int gr = m0 + wm + i * 16 + v + 8 * kh;
                const int gc = n0 + wn + j * 16 + lr;
                Out[(size_t)gr * OUTF + gc] = c[i][j][v];
            }
}

// -------------- deterministic column partial sums (no atomics) -------------
__global__ __launch_bounds__(256)
void bcm_colsum_part(const float* __restrict__ Out, float* __restrict__ Partial)
{
    const int col = blockIdx.x * 256 + threadIdx.x;     // 0..4095
    const int r0  = blockIdx.y * (BATCH / 8);           // 8 row-slabs of 1024
    float s = 0.0f;
#pragma unroll 4
    for (int r = 0; r < BATCH / 8; ++r)
        s += Out[(size_t)(r0 + r) * OUTF + col];
    Partial[blockIdx.y * OUTF + col] = s;
}

__global__ __launch_bounds__(256)
void bcm_threshold(const float* __restrict__ Partial, const float* __restrict__ Th,
                   float* __restrict__ NewTh)
{
    const int i = blockIdx.x * 256 + threadIdx.x;
    float s = 0.0f;
#pragma unroll
    for (int p = 0; p < 8; ++p) s += Partial[p * OUTF + i];
    const float mean = s * (1.0f / (float)BATCH);
    const float th   = Th[i];
    NewTh[i] = th + (mean * mean - th) * (1.0f / 1000.0f);   // TAU = 1000
}

// ---------- GEMM2: W' = W + (LR/B) * relu(out - th')^T @ x -----------------
__global__ __launch_bounds__(256)
void bcm_gemm_upd(const float* __restrict__ Out, const float* __restrict__ X,
                  const float* __restrict__ NewTh, const float* __restrict__ Wold,
                  float* __restrict__ Wnew)
{
    __shared__ float sA[2][TILE_K * LDS_M_STRIDE]; // out^T tile, [k][m] (raw copy)
    __shared__ float sB[2][TILE_K * LDS_M_STRIDE]; // x tile,     [k][n]

    const int t    = threadIdx.x;
    const int m0   = blockIdx.y * TILE_M;   // out-feature tile
    const int n0   = blockIdx.x * TILE_N;   // in-feature tile
    const int wave = t >> 5;
    const int lane = t & 31;
    const int kh   = (lane >> 4) & 1;
    const int lr   = lane & 15;
    const int wm   = (wave >> 2) * 64;
    const int wn   = (wave & 3) * 32;

    const int lkrow = t >> 3;               // 0..31 (k rows)
    const int lcol  = (t & 7) * 16;         // 0..112 (column base)

    const uint32_t ldsA[2] = { lds_off(&sA[0][lkrow * LDS_M_STRIDE + lcol]),
                               lds_off(&sA[1][lkrow * LDS_M_STRIDE + lcol]) };
    const uint32_t ldsB[2] = { lds_off(&sB[0][lkrow * LDS_M_STRIDE + lcol]),
                               lds_off(&sB[1][lkrow * LDS_M_STRIDE + lcol]) };
    const uint32_t voffA = (uint32_t)(lkrow * OUTF + m0 + lcol) * 4u;
    const uint32_t voffB = (uint32_t)(lkrow * INF  + n0 + lcol) * 4u;

    // per-lane thresholds: a-frag (x,y) share the same output row M
    float tha[4];
#pragma unroll
    for (int i = 0; i < 4; ++i) tha[i] = NewTh[m0 + wm + i * 16 + lr];

    v8f c[4][2];
#pragma unroll
    for (int i = 0; i < 4; ++i)
#pragma unroll
        for (int j = 0; j < 2; ++j) c[i][j] = (v8f)0.0f;

    ASYNC_TILE_B128(ldsA[0], voffA, Out);
    ASYNC_TILE_B128(ldsB[0], voffB, X);
    WAIT_ASYNC0();
    __syncthreads();

    for (int k0 = 0; k0 < BATCH; k0 += TILE_K) {
        const int cur = (k0 / TILE_K) & 1;

        if (k0 + TILE_K < BATCH) {
            ASYNC_TILE_B128(ldsA[cur ^ 1], voffA, Out + (size_t)(k0 + TILE_K) * OUTF);
            ASYNC_TILE_B128(ldsB[cur ^ 1], voffB, X   + (size_t)(k0 + TILE_K) * INF);
        }

        const float* __restrict__ curA = sA[cur];
        const float* __restrict__ curB = sB[cur];
#pragma unroll
        for (int kk = 0; kk < TILE_K; kk += 4) {
            const int ko = kk + 2 * kh;
            v2f a[4], b[2];
#pragma unroll
            for (int i = 0; i < 4; ++i) {
                const int mi = wm + i * 16 + lr;
                a[i].x = fmaxf(curA[ko * LDS_M_STRIDE + mi]       - tha[i], 0.0f);
                a[i].y = fmaxf(curA[(ko + 1) * LDS_M_STRIDE + mi] - tha[i], 0.0f);
            }
#pragma unroll
            for (int j = 0; j < 2; ++j) {
                const int nj = wn + j * 16 + lr;
                b[j].x = curB[ko * LDS_M_STRIDE + nj];
                b[j].y = curB[(ko + 1) * LDS_M_STRIDE + nj];
            }
#pragma unroll
            for (int i = 0; i < 4; ++i)
#pragma unroll
                for (int j = 0; j < 2; ++j)
                    c[i][j] = __builtin_amdgcn_wmma_f32_16x16x4_f32(
                        false, a[i], false, b[j], (short)0, c[i][j], false, false);
        }

        WAIT_ASYNC0();
        __syncthreads();
    }

    const float scale = 0.001f / (float)BATCH;   // LR / B
#pragma unroll
    for (int i = 0; i < 4; ++i)
#pragma unroll
        for (int j = 0; j < 2; ++j)
#pragma unroll
            for (int v = 0; v < 8; ++v) {
                const int gr = m0 + wm + i * 16 + v + 8 * kh;
                const int gc = n0 + wn + j * 16 + lr;
                const size_t idx = (size_t)gr * INF + gc;
                Wnew[idx] = Wold[idx] + scale * c[i][j][v];
            }
}

// ---------------------------------------------------------------------------
extern "C" void kernel_launch(void* const* d_in, const int* in_sizes, int n_in,
                              void* d_out, int out_size, void* d_ws, size_t ws_size,
                              hipStream_t stream) {
    (void)in_sizes; (void)n_in; (void)out_size; (void)ws_size;

    const float* x  = (const float*)d_in[0];   // (8192, 4096)
    const float* w  = (const float*)d_in[1];   // (4096, 4096)
    const float* th = (const float*)d_in[2];   // (4096,)

    float* out   = (float*)d_out;                        // (8192, 4096)
    float* newTh = out + (size_t)BATCH * OUTF;           // (4096,)
    float* newW  = newTh + OUTF;                         // (4096, 4096)
    float* part  = (float*)d_ws;                         // 8 x 4096 partial sums

    // 1) forward GEMM: out = x @ W^T
    bcm_gemm_fwd<<<dim3(OUTF / TILE_N, BATCH / TILE_M), 256, 0, stream>>>(x, w, out);

    // 2) deterministic column means -> threshold EMA
    bcm_colsum_part<<<dim3(OUTF / 256, 8), 256, 0, stream>>>(out, part);
    bcm_threshold<<<dim3(OUTF / 256), 256, 0, stream>>>(part, th, newTh);

    // 3) Hebbian update GEMM: W' = W + (LR/B) * relu(out - th')^T @ x
    bcm_gemm_upd<<<dim3(INF / TILE_N, OUTF / TILE_M), 256, 0, stream>>>(
        out, x, newTh, w, newW);
}